// PPNProp_3178275799578
// MI455X (gfx1250) — compile-verified
//
#include <hip/hip_runtime.h>
#include <stdint.h>

// PPNP propagation: deterministic edge dropout + SpMM scatter.
// MI455X reasoning: x (25.6 MB) and out (25.6 MB) both fit in the 192 MB L2,
// so the random gather/scatter stays on-chip; only the 19.2 MB edge stream
// touches HBM (~1 us at 23.3 TB/s). AI ~= 0.25 FLOP/B -> pure memory problem;
// WMMA intentionally unused. CDNA5 paths used: async global->LDS staging of
// the edge stream (ASYNCcnt double-buffer), TH_LOAD_NT_RT gather hints,
// lane-parallel global_prefetch_b8, L2-resident f32 atomics.

#define N_FEAT 128
#define CHUNK 32                       // edges staged per async op
#define KEEP_THRESH 15099494u          // floor(0.9 * 2^24)
#define INV_KEEP 1.1111111111111112f   // 1 / 0.9

__global__ void __launch_bounds__(256)
zero_out_kernel(float4* __restrict__ out, int n4) {
    int i = blockIdx.x * blockDim.x + threadIdx.x;
    const int stride = gridDim.x * blockDim.x;
    const float4 z = make_float4(0.f, 0.f, 0.f, 0.f);
    for (; i < n4; i += stride) out[i] = z;
}

__device__ __forceinline__ uint32_t edge_hash(uint32_t x) {
    // fixed-seed integer hash -> deterministic bernoulli(0.9) keep mask
    x ^= 0x2545F491u;
    x ^= x >> 16; x *= 0x7feb352du;
    x ^= x >> 15; x *= 0x846ca68bu;
    x ^= x >> 16;
    return x;
}

// One edge: weighted gather of a 128-feature row + scatter-add.
// TH_LOAD_NT_RT: non-temporal in WGP$/L0 (random gather thrashes it),
// regular-temporal in L2 where x is resident.
__device__ __forceinline__ void process_edge(const float* __restrict__ x,
                                             float* __restrict__ out,
                                             int lane, uint32_t e,
                                             float v_raw, int r, int c) {
    const uint32_t h = edge_hash(e);
    const float v = ((h >> 8) < KEEP_THRESH) ? v_raw * INV_KEEP : 0.0f;
    if (v == 0.0f) return;            // dropped edge: skip 1 KB of L2 traffic

    const float* xp = x + (size_t)c * N_FEAT + (size_t)lane * 4;
    float4 xv;
    asm volatile("global_load_b128 %0, %1, off th:TH_LOAD_NT_RT\n\t"
                 "s_wait_loadcnt 0x0"
                 : "=v"(xv) : "v"(xp) : "memory");

    float* o = out + (size_t)r * N_FEAT + (size_t)lane * 4;
    atomicAdd(o + 0, v * xv.x);
    atomicAdd(o + 1, v * xv.y);
    atomicAdd(o + 2, v * xv.z);
    atomicAdd(o + 3, v * xv.w);
}

__global__ void __launch_bounds__(256)
spmm_scatter_kernel(const float* __restrict__ x,
                    const float* __restrict__ vals,
                    const int* __restrict__ row,
                    const int* __restrict__ col,
                    float* __restrict__ out,
                    int n_edges) {
    // 8 waves * 2 buffers * 384 B (32 edges: 128B vals | 128B row | 128B col)
    __shared__ uint32_t smem[8 * 2 * 96];

    const int lane   = threadIdx.x & 31;
    const int wib    = __builtin_amdgcn_readfirstlane((int)(threadIdx.x >> 5));
    const int wpb    = (int)(blockDim.x >> 5);
    const int gwid   = (int)blockIdx.x * wpb + wib;     // global wave id
    const int nwaves = (int)gridDim.x * wpb;

    const int n_full   = n_edges & ~(CHUNK - 1);
    const int n_chunks = n_full / CHUNK;

    // Async routing: one b128 per lane stages the whole 384 B chunk in ONE op.
    // lanes 0-7 -> vals, 8-15 -> row, 16-23 -> col, 24-31 idle (EXEC-masked).
    const int grp = lane >> 3;
    const int sub = lane & 7;
    const char* abase = (grp == 0) ? (const char*)vals
                      : (grp == 1) ? (const char*)row
                                   : (const char*)col;

    uint32_t* wbuf0 = smem + wib * 192;                 // this wave's LDS region
    const uint32_t wave_lds = (uint32_t)(size_t)wbuf0;  // LDS byte offset
    const uint32_t lane_dst = wave_lds + (uint32_t)grp * 128u + (uint32_t)sub * 16u;

    int cid = gwid;
    int buf = 0;

    // Prologue: stage first chunk (buffer 0)
    if (cid < n_chunks && grp < 3) {
        const char* g = abase + ((size_t)cid * CHUNK + (size_t)sub * 4) * 4;
        asm volatile("global_load_async_to_lds_b128 %0, %1, off"
                     :: "v"(lane_dst), "v"(g) : "memory");
    }

    for (; cid < n_chunks; cid += nwaves) {
        const int ncid = cid + nwaves;
        const int nbuf = buf ^ 1;
        if (ncid < n_chunks) {
            // Stage next chunk into the other buffer, then wait for current:
            // async loads complete in order, so ASYNCcnt<=1 (newest may remain
            // outstanding) guarantees the current chunk has landed in LDS.
            if (grp < 3) {
                const char* g = abase + ((size_t)ncid * CHUNK + (size_t)sub * 4) * 4;
                const uint32_t d = lane_dst + (uint32_t)nbuf * 384u;
                asm volatile("global_load_async_to_lds_b128 %0, %1, off"
                             :: "v"(d), "v"(g) : "memory");
            }
            asm volatile("s_wait_asynccnt 0x1" ::: "memory");
        } else {
            asm volatile("s_wait_asynccnt 0x0" ::: "memory");
        }

        uint32_t* wb = wbuf0 + buf * 96;   // [0..31]=vals [32..63]=row [64..95]=col

        // Lane-parallel prefetch: one global_prefetch_b8 warms the first
        // cacheline of all 32 gather rows of this chunk at once.
        {
            const int c_pf = (int)wb[64 + lane];
            __builtin_prefetch(x + (size_t)c_pf * N_FEAT, 0, 3);
        }

        const uint32_t ebase = (uint32_t)cid * CHUNK;
        #pragma unroll 4
        for (int k = 0; k < CHUNK; ++k) {
            const float v_raw = ((const float*)wb)[k];   // LDS broadcast reads
            const int   r     = (int)wb[32 + k];
            const int   c     = (int)wb[64 + k];
            process_edge(x, out, lane, ebase + (uint32_t)k, v_raw, r, c);
        }
        buf = nbuf;
    }

    // Generic tail (n_edges not divisible by CHUNK): one edge per wave,
    // direct scalar-path loads. Empty for the 1.6M-edge reference shape.
    {
        const int e = n_full + gwid;
        if (e < n_edges) {
            const float v_raw = __builtin_nontemporal_load(vals + e);
            const int   r     = __builtin_nontemporal_load(row + e);
            const int   c     = __builtin_nontemporal_load(col + e);
            process_edge(x, out, lane, (uint32_t)e, v_raw, r, c);
        }
    }
}

extern "C" void kernel_launch(void* const* d_in, const int* in_sizes, int n_in,
                              void* d_out, int out_size, void* d_ws, size_t ws_size,
                              hipStream_t stream) {
    const float* x    = (const float*)d_in[0];
    const float* vals = (const float*)d_in[1];
    const int*   row  = (const int*)d_in[2];
    const int*   col  = (const int*)d_in[3];
    float*       out  = (float*)d_out;

    const int n_edges = in_sizes[1];
    const int n4      = out_size / 4;   // out zeroed as float4

    // 1) zero the (poisoned) output accumulator
    zero_out_kernel<<<4096, 256, 0, stream>>>((float4*)out, n4);

    // 2) chunked edge scatter: 2048 blocks * 8 waves = 16384 waves,
    //    ~3 chunks (96 edges) per wave -> async double-buffer stays busy.
    spmm_scatter_kernel<<<2048, 256, 0, stream>>>(x, vals, row, col, out, n_edges);
}